// STGridModule_28578712388215
// MI455X (gfx1250) — compile-verified
//
#include <hip/hip_runtime.h>
#include <hip/hip_bf16.h>

// ---------------------------------------------------------------------------
// Problem: T=512, S=512, F=256, f32 throughout.
//   s_pool[t,f] = max_s x[t,s,f]     (512x256)
//   t_pool[s,f] = max_t x[t,s,f]     (512x256)
//   s_proj[t,g] = sum_f s_pool[t,f]*W[g,f]        (W1 = W[:, :256], W row stride 512)
//   t_proj[s,g] = sum_f t_pool[s,f]*W[g,256+f]
//   out[t,s,g]  = s_proj[t,g] + t_proj[s,g] + b[g]
// Bandwidth-bound: ~512 MiB essential traffic -> ~22us @ 23.3 TB/s.
// GEMMs (134 MFLOP) done with V_WMMA_F32_16X16X4_F32 (full f32 precision).
// ---------------------------------------------------------------------------

typedef __attribute__((ext_vector_type(2))) float v2f;
typedef __attribute__((ext_vector_type(4))) float f4;
typedef __attribute__((ext_vector_type(8))) float v8f;

#define TT 512
#define SS 512
#define FF 256
#define NBLK 16   // 512/32 partial blocks per reduced dimension

// ---------------------------------------------------------------------------
// Kernel 1: single pass over x computes partial maxes for BOTH pools.
// Grid (16,16): blockIdx.x = s-block, blockIdx.y = t-block, 32x32 (t,s) tile.
// 256 threads = one f column per lane -> 128B contiguous per-wave loads.
// ps[t][sblk][f] : partial over 32 s values;  pt[s][tblk][f] : partial over 32 t.
// ---------------------------------------------------------------------------
__global__ __launch_bounds__(256) void pool_partial(const float* __restrict__ x,
                                                    float* __restrict__ ps,
                                                    float* __restrict__ pt) {
  const int f    = threadIdx.x;        // 0..255
  const int sblk = blockIdx.x;         // 0..15
  const int tblk = blockIdx.y;         // 0..15
  const int s0   = sblk * 32;
  const int t0   = tblk * 32;

  float tp[32];
#pragma unroll
  for (int s = 0; s < 32; ++s) tp[s] = -__builtin_inff();

  for (int t = 0; t < 32; ++t) {
    const float* row = x + ((size_t)(t0 + t) * SS + s0) * FF + f;
    float smax = -__builtin_inff();
#pragma unroll
    for (int s = 0; s < 32; ++s) {             // 32 independent loads -> MLP
      float v = row[(size_t)s * FF];
      smax  = fmaxf(smax, v);
      tp[s] = fmaxf(tp[s], v);
    }
    ps[((size_t)(t0 + t) * NBLK + sblk) * FF + f] = smax;
  }
#pragma unroll
  for (int s = 0; s < 32; ++s)
    pt[((size_t)(s0 + s) * NBLK + tblk) * FF + f] = tp[s];
}

// ---------------------------------------------------------------------------
// Kernel 2: reduce the 16 partials per row. Blocks 0..511 -> s_pool,
// blocks 512..1023 -> t_pool. One f per lane, coalesced.
// ---------------------------------------------------------------------------
__global__ __launch_bounds__(256) void pool_reduce(const float* __restrict__ ps,
                                                   const float* __restrict__ pt,
                                                   float* __restrict__ s_pool,
                                                   float* __restrict__ t_pool) {
  const int f = threadIdx.x;
  int r = blockIdx.x;                  // 0..1023
  const float* src;
  float* dst;
  if (r < TT) { src = ps + (size_t)r * NBLK * FF;          dst = s_pool + (size_t)r * FF; }
  else        { r -= TT; src = pt + (size_t)r * NBLK * FF; dst = t_pool + (size_t)r * FF; }
  float m = -__builtin_inff();
#pragma unroll
  for (int i = 0; i < NBLK; ++i) m = fmaxf(m, src[(size_t)i * FF + f]);
  dst[f] = m;
}

// ---------------------------------------------------------------------------
// Kernel 3: both projections via V_WMMA_F32_16X16X4_F32 (full f32 precision).
// 1024 16x16 output tiles (rows 0..511 = s_proj, 512..1023 = t_proj),
// one wave per tile, 8 waves per block, 128 blocks.
// ISA fragment layouts (wave32, h = lane>=16):
//   A 16x4 : a.x = A[lane&15][k+2h],   a.y = A[lane&15][k+2h+1]
//   B 4x16 : b.x = B[k+2h][lane&15] = Wsub[lane&15][k+2h]  (einsum uses W[g,f])
//   C/D    : c[r] -> row r+8h, col lane&15
// ---------------------------------------------------------------------------
__global__ __launch_bounds__(256) void gemm_wmma(const float* __restrict__ s_pool,
                                                 const float* __restrict__ t_pool,
                                                 const float* __restrict__ W,
                                                 float* __restrict__ s_proj,
                                                 float* __restrict__ t_proj) {
  const int wave = threadIdx.x >> 5;
  const int lane = threadIdx.x & 31;
  const int tile = blockIdx.x * 8 + wave;   // 0..1023
  const int mt   = tile >> 4;               // 0..63 (row tile)
  const int nt   = tile & 15;               // 0..15 (col tile)

  const float* A;
  float* out;
  int off, row0 = mt * 16;
  if (row0 < TT) { A = s_pool; out = s_proj; off = 0; }
  else           { row0 -= TT; A = t_pool; out = t_proj; off = FF; }

  const int h = lane >> 4;                  // 0 or 1
  const int r = lane & 15;

  const float* aptr = A + (size_t)(row0 + r) * FF + 2 * h;
  const float* bptr = W + (size_t)(nt * 16 + r) * (2 * FF) + off + 2 * h;

  v8f c = {};
#pragma unroll 8
  for (int k = 0; k < FF; k += 4) {
    v2f a = *(const v2f*)(aptr + k);
    v2f b = *(const v2f*)(bptr + k);
    // 8 args: (neg_a, A, neg_b, B, c_mod, C, reuse_a, reuse_b)
    c = __builtin_amdgcn_wmma_f32_16x16x4_f32(false, a, false, b, (short)0, c,
                                              false, false);
  }

  float* o = out + (size_t)row0 * FF + nt * 16 + r;
#pragma unroll
  for (int rr = 0; rr < 8; ++rr)
    o[(size_t)(rr + 8 * h) * FF] = c[rr];
}

// ---------------------------------------------------------------------------
// Kernel 4: out[t,s,g] = s_proj[t,g] + t_proj[s,g] + b[g].  float4 lanes,
// g-fastest -> 512B contiguous wave stores; nontemporal (never re-read).
// Grid 8192 blocks x 256 threads; each thread: fixed (t, g4), 8 s values.
// ---------------------------------------------------------------------------
__global__ __launch_bounds__(256) void bcast_add(const float* __restrict__ s_proj,
                                                 const float* __restrict__ t_proj,
                                                 const float* __restrict__ b,
                                                 float* __restrict__ out) {
  const int g4     = threadIdx.x & 63;        // float4 index within F (64 * 4 = 256)
  const int so     = threadIdx.x >> 6;        // 0..3
  const int t      = blockIdx.x >> 4;         // 0..511
  const int s_base = (blockIdx.x & 15) * 32 + so * 8;

  const f4* sp4 = (const f4*)s_proj;
  const f4* tp4 = (const f4*)t_proj;
  const f4* b4  = (const f4*)b;

  const f4 base = sp4[(size_t)t * 64 + g4] + b4[g4];
  f4* o = (f4*)out + ((size_t)t * SS + s_base) * 64 + g4;

#pragma unroll
  for (int i = 0; i < 8; ++i) {
    f4 tv = tp4[(size_t)(s_base + i) * 64 + g4];
    __builtin_nontemporal_store(base + tv, o + (size_t)i * 64);
  }
}

// ---------------------------------------------------------------------------
extern "C" void kernel_launch(void* const* d_in, const int* in_sizes, int n_in,
                              void* d_out, int out_size, void* d_ws, size_t ws_size,
                              hipStream_t stream) {
  const float* x = (const float*)d_in[0];   // (512,512,256)
  const float* W = (const float*)d_in[1];   // (256,512)
  const float* b = (const float*)d_in[2];   // (256,)
  float* out = (float*)d_out;               // (512,512,256)

  // Workspace layout (floats): 2*2097152 + 4*131072 = 4,718,592 (18 MiB)
  float* ws     = (float*)d_ws;
  float* ps     = ws;                        // [512][16][256]
  float* pt     = ps + (size_t)TT * NBLK * FF;
  float* s_pool = pt + (size_t)SS * NBLK * FF;
  float* t_pool = s_pool + (size_t)TT * FF;
  float* s_proj = t_pool + (size_t)SS * FF;
  float* t_proj = s_proj + (size_t)TT * FF;

  pool_partial<<<dim3(16, 16), 256, 0, stream>>>(x, ps, pt);
  pool_reduce<<<1024, 256, 0, stream>>>(ps, pt, s_pool, t_pool);
  gemm_wmma<<<128, 256, 0, stream>>>(s_pool, t_pool, W, s_proj, t_proj);
  bcast_add<<<8192, 256, 0, stream>>>(s_proj, t_proj, b, out);
}